// EMDLoss_65987877536025
// MI455X (gfx1250) — compile-verified
//
#include <hip/hip_runtime.h>
#include <math.h>

#define BB    16
#define NN    2048
#define BN    (BB * NN)
#define EPSV  0.005f
#define NITER 50
#define LOGW  (-7.62461899f)   /* -ln(2048) */
#define LOG2E 1.44269504f
#define LN2   0.69314718f

typedef float v2f __attribute__((ext_vector_type(2)));
typedef float v8f __attribute__((ext_vector_type(8)));

__device__ __forceinline__ v8f wmma4(v2f a, v2f b, v8f c) {
  // V_WMMA_F32_16X16X4_F32 : D = A(16x4) * B(4x16) + C(16x16), fp32
  return __builtin_amdgcn_wmma_f32_16x16x4_f32(false, a, false, b, (short)0, c,
                                               false, false);
}

// ---------------------------------------------------------------------------
// Preprocess 1: per-batch min (offset) over pred∪target, then per-batch max of
// (coord - offset). One block per batch.
// ---------------------------------------------------------------------------
__global__ __launch_bounds__(256) void pre_minmax(
    const float* __restrict__ pred, const float* __restrict__ target,
    float* __restrict__ offs, float* __restrict__ bmax) {
  const int b = blockIdx.x;
  const int tid = threadIdx.x;
  const int lane = tid & 31, wave = tid >> 5;
  __shared__ float red[8][3];
  __shared__ float roff[3];

  float mn0 = 3.0e38f, mn1 = 3.0e38f, mn2 = 3.0e38f;
  for (int i = tid; i < 2 * NN; i += 256) {
    const float* src = (i < NN) ? (pred + (size_t)(b * NN + i) * 3)
                                : (target + (size_t)(b * NN + i - NN) * 3);
    mn0 = fminf(mn0, src[0]);
    mn1 = fminf(mn1, src[1]);
    mn2 = fminf(mn2, src[2]);
  }
#pragma unroll
  for (int m = 1; m < 32; m <<= 1) {
    mn0 = fminf(mn0, __shfl_xor(mn0, m, 32));
    mn1 = fminf(mn1, __shfl_xor(mn1, m, 32));
    mn2 = fminf(mn2, __shfl_xor(mn2, m, 32));
  }
  if (lane == 0) { red[wave][0] = mn0; red[wave][1] = mn1; red[wave][2] = mn2; }
  __syncthreads();
  if (tid == 0) {
    float o0 = red[0][0], o1 = red[0][1], o2 = red[0][2];
    for (int w = 1; w < 8; ++w) {
      o0 = fminf(o0, red[w][0]);
      o1 = fminf(o1, red[w][1]);
      o2 = fminf(o2, red[w][2]);
    }
    offs[b * 3 + 0] = o0; offs[b * 3 + 1] = o1; offs[b * 3 + 2] = o2;
    roff[0] = o0; roff[1] = o1; roff[2] = o2;
  }
  __syncthreads();
  const float o0 = roff[0], o1 = roff[1], o2 = roff[2];
  float mx = -3.0e38f;
  for (int i = tid; i < 2 * NN; i += 256) {
    const float* src = (i < NN) ? (pred + (size_t)(b * NN + i) * 3)
                                : (target + (size_t)(b * NN + i - NN) * 3);
    mx = fmaxf(mx, src[0] - o0);
    mx = fmaxf(mx, src[1] - o1);
    mx = fmaxf(mx, src[2] - o2);
  }
#pragma unroll
  for (int m = 1; m < 32; m <<= 1) mx = fmaxf(mx, __shfl_xor(mx, m, 32));
  if (lane == 0) red[wave][0] = mx;
  __syncthreads();
  if (tid == 0) {
    float v = red[0][0];
    for (int w = 1; w < 8; ++w) v = fmaxf(v, red[w][0]);
    bmax[b] = v;
  }
}

// ---------------------------------------------------------------------------
// Preprocess 2: global scale = 0.99999 / max_b bmax[b]
// ---------------------------------------------------------------------------
__global__ void pre_scale(const float* __restrict__ bmax,
                          float* __restrict__ scale) {
  const int lane = threadIdx.x;
  float v = (lane < BB) ? bmax[lane] : -3.0e38f;
#pragma unroll
  for (int m = 1; m < 32; m <<= 1) v = fmaxf(v, __shfl_xor(v, m, 32));
  if (lane == 0) scale[0] = 0.99999f / v;
}

// ---------------------------------------------------------------------------
// Preprocess 3: scaled SoA points + squared norms, init f = g = 0.
// ---------------------------------------------------------------------------
__global__ __launch_bounds__(256) void pre_pack(
    const float* __restrict__ pred, const float* __restrict__ target,
    const float* __restrict__ offs, const float* __restrict__ scale,
    float* __restrict__ PX, float* __restrict__ PY, float* __restrict__ PZ,
    float* __restrict__ NP, float* __restrict__ TX, float* __restrict__ TY,
    float* __restrict__ TZ, float* __restrict__ NT, float* __restrict__ F,
    float* __restrict__ G) {
  const int i = blockIdx.x * 256 + threadIdx.x;
  const int b = i >> 11;  // i / NN
  const float sc = scale[0];
  const float o0 = offs[b * 3 + 0], o1 = offs[b * 3 + 1], o2 = offs[b * 3 + 2];
  float px = (pred[(size_t)i * 3 + 0] - o0) * sc;
  float py = (pred[(size_t)i * 3 + 1] - o1) * sc;
  float pz = (pred[(size_t)i * 3 + 2] - o2) * sc;
  PX[i] = px; PY[i] = py; PZ[i] = pz;
  NP[i] = px * px + py * py + pz * pz;
  float tx = (target[(size_t)i * 3 + 0] - o0) * sc;
  float ty = (target[(size_t)i * 3 + 1] - o1) * sc;
  float tz = (target[(size_t)i * 3 + 2] - o2) * sc;
  TX[i] = tx; TY[i] = ty; TZ[i] = tz;
  NT[i] = tx * tx + ty * ty + tz * tz;
  F[i] = 0.0f;
  G[i] = 0.0f;
}

// ---------------------------------------------------------------------------
// One Sinkhorn half-update:  out_i = eps*(logw - LSE_j((dual_j - C_ij)/eps))
// D'_ij = A_i . B_j is the BASE-2 exponent (log2e folded into B):
//   A_i = [rx, ry, rz, 1]
//   B_j = log2e/eps * [2cx, 2cy, 2cz, (dual_j - cn_j)]
// LSE_nat = (M' + log2 S) * ln2 ;  out_i = eps*(logw - LSE_nat) + rn_i.
// Inner loop: 1 WMMA per 16x16 tile + raw v_exp_f32 per element (no mul).
// Grid: (NN/128, BB); block 256 = 8 waves, each wave owns a 16-row strip.
// ---------------------------------------------------------------------------
__global__ __launch_bounds__(256) void sinkhorn_update(
    const float* __restrict__ rx, const float* __restrict__ ry,
    const float* __restrict__ rz, const float* __restrict__ rn,
    const float* __restrict__ cx, const float* __restrict__ cy,
    const float* __restrict__ cz, const float* __restrict__ cn,
    const float* __restrict__ dualc, float* __restrict__ out) {
  __shared__ float sB[4][NN];
  const int b = blockIdx.y;
  const int base = b * NN;
  const int tid = threadIdx.x;
  const float k2 = 2.0f * LOG2E / EPSV;  // coord scaling (base-2 exponent)
  const float k1 = LOG2E / EPSV;

  for (int j = tid; j < NN; j += 256) {
    sB[0][j] = cx[base + j] * k2;
    sB[1][j] = cy[base + j] * k2;
    sB[2][j] = cz[base + j] * k2;
    sB[3][j] = (dualc[base + j] - cn[base + j]) * k1;
  }
  __syncthreads();

  const int wave = tid >> 5, lane = tid & 31;
  const int half = lane >> 4, col = lane & 15;
  const int i0 = base + blockIdx.x * 128 + wave * 16;

  // A operand per ISA layout: lanes 0-15 K=0/1, lanes 16-31 K=2/3, M=col
  v2f a;
  a.x = half ? rz[i0 + col] : rx[i0 + col];
  a.y = half ? 1.0f : ry[i0 + col];
  const float* pb0 = half ? &sB[2][col] : &sB[0][col];
  const float* pb1 = half ? &sB[3][col] : &sB[1][col];

  // Phase A: per-row max of D' over all j (per-lane, then 16-lane butterfly)
  float mrow[8];
#pragma unroll
  for (int r = 0; r < 8; ++r) mrow[r] = -3.0e38f;
  for (int j0 = 0; j0 < NN; j0 += 16) {
    v2f bb; bb.x = pb0[j0]; bb.y = pb1[j0];
    v8f acc = {};
    acc = wmma4(a, bb, acc);
#pragma unroll
    for (int r = 0; r < 8; ++r) mrow[r] = fmaxf(mrow[r], acc[r]);
  }
#pragma unroll
  for (int r = 0; r < 8; ++r) {
#pragma unroll
    for (int m = 1; m < 16; m <<= 1)
      mrow[r] = fmaxf(mrow[r], __shfl_xor(mrow[r], m, 16));
  }

  // Phase B: S_i = sum_j 2^(D'_ij - M'_i)  -- raw v_exp_f32, no scaling mul
  float srow[8];
#pragma unroll
  for (int r = 0; r < 8; ++r) srow[r] = 0.0f;
  for (int j0 = 0; j0 < NN; j0 += 16) {
    v2f bb; bb.x = pb0[j0]; bb.y = pb1[j0];
    v8f acc = {};
    acc = wmma4(a, bb, acc);
#pragma unroll
    for (int r = 0; r < 8; ++r)
      srow[r] += __builtin_amdgcn_exp2f(acc[r] - mrow[r]);
  }
#pragma unroll
  for (int r = 0; r < 8; ++r) {
#pragma unroll
    for (int m = 1; m < 16; m <<= 1) srow[r] += __shfl_xor(srow[r], m, 16);
  }

  if (col == 0) {
#pragma unroll
    for (int r = 0; r < 8; ++r) {
      const int row = i0 + half * 8 + r;
      const float lse_nat =
          (mrow[r] + __builtin_amdgcn_logf(srow[r])) * LN2;  // v_log_f32=log2
      out[row] = EPSV * (LOGW - lse_nat) + rn[row];
    }
  }
}

// ---------------------------------------------------------------------------
// Finalize: assign_i = argmax_j D'_ij (== argmax_j (g_j - C_ij), monotone),
// dis = ||p_i - t_assign||^2 ; per-block partial sum of sqrt(dis).
// ---------------------------------------------------------------------------
__global__ __launch_bounds__(256) void emd_finalize(
    const float* __restrict__ rx, const float* __restrict__ ry,
    const float* __restrict__ rz, const float* __restrict__ cx,
    const float* __restrict__ cy, const float* __restrict__ cz,
    const float* __restrict__ cn, const float* __restrict__ dualc,
    float* __restrict__ part) {
  __shared__ float sB[4][NN];
  __shared__ float sWave[8];
  const int b = blockIdx.y;
  const int base = b * NN;
  const int tid = threadIdx.x;
  const float k2 = 2.0f * LOG2E / EPSV;
  const float k1 = LOG2E / EPSV;

  for (int j = tid; j < NN; j += 256) {
    sB[0][j] = cx[base + j] * k2;
    sB[1][j] = cy[base + j] * k2;
    sB[2][j] = cz[base + j] * k2;
    sB[3][j] = (dualc[base + j] - cn[base + j]) * k1;
  }
  __syncthreads();

  const int wave = tid >> 5, lane = tid & 31;
  const int half = lane >> 4, col = lane & 15;
  const int i0 = base + blockIdx.x * 128 + wave * 16;

  v2f a;
  a.x = half ? rz[i0 + col] : rx[i0 + col];
  a.y = half ? 1.0f : ry[i0 + col];
  const float* pb0 = half ? &sB[2][col] : &sB[0][col];
  const float* pb1 = half ? &sB[3][col] : &sB[1][col];

  float bv[8];
  int bj[8];
#pragma unroll
  for (int r = 0; r < 8; ++r) { bv[r] = -3.0e38f; bj[r] = 0; }
  for (int j0 = 0; j0 < NN; j0 += 16) {
    v2f bb; bb.x = pb0[j0]; bb.y = pb1[j0];
    v8f acc = {};
    acc = wmma4(a, bb, acc);
#pragma unroll
    for (int r = 0; r < 8; ++r) {
      const bool take = acc[r] > bv[r];
      bv[r] = take ? acc[r] : bv[r];
      bj[r] = take ? (j0 + col) : bj[r];
    }
  }
#pragma unroll
  for (int r = 0; r < 8; ++r) {
#pragma unroll
    for (int m = 1; m < 16; m <<= 1) {
      const float ov = __shfl_xor(bv[r], m, 16);
      const int oj = __shfl_xor(bj[r], m, 16);
      const bool take = (ov > bv[r]) || (ov == bv[r] && oj < bj[r]);
      bv[r] = take ? ov : bv[r];
      bj[r] = take ? oj : bj[r];
    }
  }

  float val = 0.0f;
  if (col < 8) {
    const int r = col;
    const int row = i0 + half * 8 + r;
    const int j = bj[r];
    const float kt = EPSV * 0.5f * LN2;  // undo 2*log2e/eps on stored coords
    const float tx = sB[0][j] * kt;
    const float ty = sB[1][j] * kt;
    const float tz = sB[2][j] * kt;
    const float dx = rx[row] - tx, dy = ry[row] - ty, dz = rz[row] - tz;
    val = sqrtf(dx * dx + dy * dy + dz * dz);
  }
#pragma unroll
  for (int m = 1; m < 32; m <<= 1) val += __shfl_xor(val, m, 32);
  if (lane == 0) sWave[wave] = val;
  __syncthreads();
  if (tid == 0) {
    float s = 0.0f;
    for (int w = 0; w < 8; ++w) s += sWave[w];
    part[blockIdx.y * gridDim.x + blockIdx.x] = s;
  }
}

// ---------------------------------------------------------------------------
// Final deterministic reduction of 256 partials -> mean(sqrt(dis)) / scale
// ---------------------------------------------------------------------------
__global__ __launch_bounds__(256) void final_reduce(
    const float* __restrict__ part, const float* __restrict__ scale,
    float* __restrict__ out) {
  __shared__ float sw[8];
  const int tid = threadIdx.x;
  const int lane = tid & 31, wave = tid >> 5;
  float v = part[tid];
#pragma unroll
  for (int m = 1; m < 32; m <<= 1) v += __shfl_xor(v, m, 32);
  if (lane == 0) sw[wave] = v;
  __syncthreads();
  if (tid == 0) {
    float s = 0.0f;
    for (int w = 0; w < 8; ++w) s += sw[w];
    out[0] = s / (float)BN / scale[0];
  }
}

extern "C" void kernel_launch(void* const* d_in, const int* in_sizes, int n_in,
                              void* d_out, int out_size, void* d_ws,
                              size_t ws_size, hipStream_t stream) {
  (void)in_sizes; (void)n_in; (void)out_size; (void)ws_size;
  const float* pred = (const float*)d_in[0];
  const float* target = (const float*)d_in[1];
  // d_in[2] (batch) is implied by layout (sorted, equal clouds) -> unused.

  float* ws = (float*)d_ws;
  float* PX = ws + 0 * BN;
  float* PY = ws + 1 * BN;
  float* PZ = ws + 2 * BN;
  float* NP = ws + 3 * BN;
  float* TX = ws + 4 * BN;
  float* TY = ws + 5 * BN;
  float* TZ = ws + 6 * BN;
  float* NT = ws + 7 * BN;
  float* F  = ws + 8 * BN;
  float* G  = ws + 9 * BN;
  float* OFFS  = ws + 10 * BN;        // 48 floats
  float* BMAX  = OFFS + 64;           // 16 floats
  float* SCALE = BMAX + 32;           // 1 float
  float* PART  = SCALE + 32;          // 256 floats

  pre_minmax<<<BB, 256, 0, stream>>>(pred, target, OFFS, BMAX);
  pre_scale<<<1, 32, 0, stream>>>(BMAX, SCALE);
  pre_pack<<<BN / 256, 256, 0, stream>>>(pred, target, OFFS, SCALE, PX, PY, PZ,
                                         NP, TX, TY, TZ, NT, F, G);

  dim3 grid(NN / 128, BB);
  for (int it = 0; it < NITER; ++it) {
    // f-update: rows = p, cols = t, dual = g
    sinkhorn_update<<<grid, 256, 0, stream>>>(PX, PY, PZ, NP, TX, TY, TZ, NT,
                                              G, F);
    // g-update: rows = t, cols = p, dual = f  (cost is structurally symmetric)
    sinkhorn_update<<<grid, 256, 0, stream>>>(TX, TY, TZ, NT, PX, PY, PZ, NP,
                                              F, G);
  }

  emd_finalize<<<grid, 256, 0, stream>>>(PX, PY, PZ, TX, TY, TZ, NT, G, PART);
  final_reduce<<<1, 256, 0, stream>>>(PART, SCALE, (float*)d_out);
}